// HGCNLayer_69672959476267
// MI455X (gfx1250) — compile-verified
//
#include <hip/hip_runtime.h>
#include <hip/hip_bf16.h>
#include <stdint.h>

// ---------------------------------------------------------------------------
// HGCN mean-aggregation layer for MI455X (gfx1250, wave32).
// Memory-bound scatter/gather: h_src (51MB) + rst (26MB) + bsrc (51MB) +
// index streams all fit in the 192MB L2, so the design leans on L2-resident
// global_atomic_add_f32, 128-bit vector loads, and the CDNA5 async
// load-to-LDS path (ASYNCcnt) for edge-index staging.
// ---------------------------------------------------------------------------

#define D_FEAT 128
#define EPB    128   // edges staged per 256-thread block
#define EPW    16    // edges per wave (8 waves * 16 = 128)

#define AS1 __attribute__((address_space(1)))
#define AS3 __attribute__((address_space(3)))

typedef int v4i __attribute__((ext_vector_type(4)));

// ---- CDNA5 async global->LDS copy (tracked by ASYNCcnt) -------------------
__device__ __forceinline__ void async_copy_b128(const int* __restrict__ g, int* l) {
#if __has_builtin(__builtin_amdgcn_global_load_async_to_lds_b128)
  __builtin_amdgcn_global_load_async_to_lds_b128(
      (AS1 v4i*)g, (AS3 v4i*)l, /*offset=*/0, /*cpol=*/0);
#else
  *(int4*)l = *(const int4*)g;
#endif
}

__device__ __forceinline__ void async_wait0() {
#if __has_builtin(__builtin_amdgcn_s_wait_asynccnt)
  __builtin_amdgcn_s_wait_asynccnt(0);
#elif __has_builtin(__builtin_amdgcn_global_load_async_to_lds_b128)
  asm volatile("s_wait_asynccnt 0" ::: "memory");
#endif
}

// ---- fast no-return f32 atomic add (global_atomic_add_f32 in L2) ----------
__device__ __forceinline__ void atomic_add_f32(float* p, float v) {
  unsafeAtomicAdd(p, v);   // hardware FP32 atomic add, no CAS loop
}

// ---------------------------------------------------------------------------
// Edge-wise weighted scatter-add (placed FIRST so disasm snippet shows it):
//   out[e_out[e]] += feat[e_in[e]] * w[e_out[e]]
// One wave (32 lanes) per edge: lane handles float4 => 128 floats/row.
// Block stages 128 edges' indices into LDS via async b128 loads first.
// ---------------------------------------------------------------------------
__global__ void __launch_bounds__(256)
k_scatter(const float4* __restrict__ feat,    // [n_in , 32] (float4 view of 128 floats)
          float*        __restrict__ out,     // [n_out, 128]
          const int*    __restrict__ e_in,    // gather index per edge
          const int*    __restrict__ e_out,   // scatter index per edge
          const float*  __restrict__ wtab,    // weight indexed by e_out
          long n_edges) {
  __shared__ int s_in [EPB];
  __shared__ int s_out[EPB];

  const long base = (long)blockIdx.x * EPB;

  // --- stage indices: waves 0 (e_in) and 1 (e_out), 32 x b128 async loads
  {
    const int t = threadIdx.x;
    if (t < 64) {
      const int  lane = t & 31;
      const int* g    = (t < 32) ? e_in : e_out;
      int*       l    = (t < 32) ? s_in : s_out;
      const long ge   = base + (long)(lane * 4);
      if (ge + 4 <= n_edges) {
        async_copy_b128(g + ge, l + lane * 4);
      } else {
#pragma unroll
        for (int k = 0; k < 4; ++k) {
          const long e = ge + k;
          l[lane * 4 + k] = (e < n_edges) ? g[e] : 0;
        }
      }
      async_wait0();   // wave-level: drain this wave's ASYNCcnt
    }
  }
  __syncthreads();

  const int  wave  = threadIdx.x >> 5;
  const int  lane  = threadIdx.x & 31;
  const long wbase = base + (long)wave * EPW;

  long rem = n_edges - wbase;
  const int cnt = rem <= 0 ? 0 : (rem >= EPW ? EPW : (int)rem);

  for (int i = 0; i < cnt; ++i) {
    const int li = wave * EPW + i;
    // Indices are wave-uniform (same LDS address across lanes); force them
    // into SGPRs so the gather becomes SADDR-form global_load_b128 and the
    // weight lookup becomes a scalar load.
    const int si = __builtin_amdgcn_readfirstlane(s_in [li]);
    const int di = __builtin_amdgcn_readfirstlane(s_out[li]);

    const float  w = wtab[di];
    const float4 v = feat[(long)si * 32 + lane];

    // prefetch next edge's row: one global_prefetch_b8 per 128B segment
    if (i + 1 < cnt && (lane & 7) == 0) {
      const int sn = __builtin_amdgcn_readfirstlane(s_in[li + 1]);
      __builtin_prefetch(&feat[(long)sn * 32 + lane], 0, 1);
    }

    float* o = out + (long)di * D_FEAT + lane * 4;
    atomic_add_f32(o + 0, v.x * w);
    atomic_add_f32(o + 1, v.y * w);
    atomic_add_f32(o + 2, v.z * w);
    atomic_add_f32(o + 3, v.w * w);
  }
}

// ---------------------------------------------------------------------------
// Zero outputs + degree counters (d_out / d_ws are poisoned 0xAA).
// ---------------------------------------------------------------------------
__global__ void __launch_bounds__(256)
k_zero(float4* __restrict__ out4, long n4, uint32_t* __restrict__ cnt, long nc) {
  const long stride = (long)gridDim.x * blockDim.x;
  long i0 = (long)blockIdx.x * blockDim.x + threadIdx.x;
  for (long i = i0; i < n4; i += stride) out4[i] = make_float4(0.f, 0.f, 0.f, 0.f);
  for (long i = i0; i < nc; i += stride) cnt[i] = 0u;
}

// ---------------------------------------------------------------------------
// Exact integer in-degree counts (both directions in one pass).
// ---------------------------------------------------------------------------
__global__ void __launch_bounds__(256)
k_degree(const int* __restrict__ e_src, const int* __restrict__ e_dst,
         uint32_t* __restrict__ deg_dst, uint32_t* __restrict__ deg_src, long n_edges) {
  long i = (long)blockIdx.x * blockDim.x + threadIdx.x;
  if (i < n_edges) {
    atomicAdd(&deg_dst[e_dst[i]], 1u);
    atomicAdd(&deg_src[e_src[i]], 1u);
  }
}

// ---------------------------------------------------------------------------
// Counts -> 1/max(deg,1) in place (norm exponent is -1).
// ---------------------------------------------------------------------------
__global__ void __launch_bounds__(256)
k_invdeg(uint32_t* __restrict__ cnt, long n) {
  long i = (long)blockIdx.x * blockDim.x + threadIdx.x;
  if (i < n) {
    uint32_t c = cnt[i];
    ((float*)cnt)[i] = 1.0f / (float)(c > 0u ? c : 1u);
  }
}

// ---------------------------------------------------------------------------
// Host launcher.
//   d_in[0] h_src f32 [n_src,128]; d_in[1] edge_src i32 [E]; d_in[2] edge_dst i32 [E]
//   d_in[3] n_dst (scalar, derived from sizes instead)
//   d_out = [ bsrc (n_src*128) | rst (n_dst*128) ]  (reference return order)
//   d_ws  = [ deg_dst u32/f32 (n_dst) | deg_src u32/f32 (n_src) ]
// ---------------------------------------------------------------------------
extern "C" void kernel_launch(void* const* d_in, const int* in_sizes, int n_in,
                              void* d_out, int out_size, void* d_ws, size_t ws_size,
                              hipStream_t stream) {
  const float* h_src = (const float*)d_in[0];
  const int*   e_src = (const int*)d_in[1];
  const int*   e_dst = (const int*)d_in[2];

  const long n_src   = (long)in_sizes[0] / D_FEAT;
  const long n_edges = (long)in_sizes[1];
  const long n_dst   = (long)out_size / D_FEAT - n_src;

  float* bsrc = (float*)d_out;
  float* rst  = (float*)d_out + n_src * D_FEAT;

  uint32_t* deg_dst = (uint32_t*)d_ws;   // [n_dst], becomes invdeg f32 in place
  uint32_t* deg_src = deg_dst + n_dst;   // [n_src], becomes invdeg f32 in place

  const long n4    = (long)out_size / 4;
  const long nodes = n_src + n_dst;

  // 1) zero outputs + counters
  k_zero<<<2048, 256, 0, stream>>>((float4*)d_out, n4, deg_dst, nodes);

  // 2) integer degrees
  k_degree<<<(unsigned)((n_edges + 255) / 256), 256, 0, stream>>>(
      e_src, e_dst, deg_dst, deg_src, n_edges);

  // 3) invert degrees in place
  k_invdeg<<<(unsigned)((nodes + 255) / 256), 256, 0, stream>>>(deg_dst, nodes);

  const unsigned nb = (unsigned)((n_edges + EPB - 1) / EPB);

  // 4) forward:  rst[d] += h_src[s] * invdeg_dst[d]
  k_scatter<<<nb, 256, 0, stream>>>((const float4*)h_src, rst,
                                    e_src, e_dst, (const float*)deg_dst, n_edges);

  // 5) backward: bsrc[s] += rst[d] * invdeg_src[s]   (reads normalized rst)
  k_scatter<<<nb, 256, 0, stream>>>((const float4*)rst, bsrc,
                                    e_dst, e_src, (const float*)deg_src, n_edges);
}